// RVQ_VAE_81595788689848
// MI455X (gfx1250) — compile-verified
//
#include <hip/hip_runtime.h>
#include <hip/hip_bf16.h>

// Problem constants (from reference)
#define BB   16
#define DD   256
#define TT   4096
#define NC   3
#define KK   512
#define NN   (BB * TT)        // 65536 residual vectors
#define MT   64               // rows (vectors) per block tile
#define ZS   260              // padded LDS row stride in floats
#define NBLK (NN / MT)        // 1024 blocks

typedef __attribute__((ext_vector_type(2))) float v2f;
typedef __attribute__((ext_vector_type(8))) float v8f;

static __device__ __forceinline__ v8f wmma_f32(v2f a, v2f b, v8f c) {
    return __builtin_amdgcn_wmma_f32_16x16x4_f32(false, a, false, b,
                                                 (short)0, c, false, false);
}

// ---------------------------------------------------------------------------
// Kernel 1: ee[c*K + k] = ||E_c[k]||^2   (3*512 rows of 256 floats)
// ---------------------------------------------------------------------------
__global__ __launch_bounds__(256) void rvq_ee_kernel(const float* __restrict__ cb,
                                                     float* __restrict__ ee) {
    int row = blockIdx.x * 256 + threadIdx.x;   // 0..1535
    if (row >= NC * KK) return;
    const float4* p = (const float4*)(cb + (size_t)row * DD);
    float s = 0.f;
#pragma unroll 8
    for (int i = 0; i < DD / 4; ++i) {
        float4 v = p[i];
        s += v.x * v.x + v.y * v.y + v.z * v.z + v.w * v.w;
    }
    ee[row] = s;
}

// ---------------------------------------------------------------------------
// Kernel 2: main RVQ kernel. One block = 64 residual vectors (one b, t-slab).
//   - stage z tile in LDS
//   - 3x: fp32-WMMA GEMM (2x2 register-blocked) vs 512 codes, per-row argmin,
//     residual update
//   - emit last-codebook indices + per-block loss partial
// ---------------------------------------------------------------------------
__global__ __launch_bounds__(256) void rvq_main_kernel(const float* __restrict__ x,
                                                       const float* __restrict__ cb,
                                                       const float* __restrict__ ee,
                                                       float* __restrict__ out_idx,
                                                       float* __restrict__ partial) {
    extern __shared__ __align__(16) unsigned char smem_raw[];
    float* zbuf    = (float*)smem_raw;            // MT*ZS floats
    float* rn      = zbuf + MT * ZS;              // MT   running ||z||^2
    float* lossrow = rn + MT;                     // MT   per-row loss accum
    int*   idxrow  = (int*)(lossrow + MT);        // MT   chosen code per row
    float* pmin    = (float*)(idxrow + MT);       // MT*4 per-(row,kq) min
    int*   pidx    = (int*)(pmin + MT * 4);       // MT*4 per-(row,kq) idx

    const int tid  = threadIdx.x;
    const int wid  = tid >> 5;
    const int lane = tid & 31;
    const int ln16 = lane & 15;
    const int hi   = lane >> 4;        // 0: K 0/1 half, 1: K 2/3 half (A/B frag)
    const int mh   = wid & 1;          // m-half: rows [mh*32, mh*32+32)
    const int kq   = wid >> 1;         // k-quarter: cols [kq*128, kq*128+128)
    const int m0   = mh * 32;

    const int mbase = blockIdx.x * MT;           // global vector index base
    const int b  = mbase >> 12;                  // / TT
    const int t0 = mbase & (TT - 1);
    const float* xb = x + (size_t)b * ((size_t)DD * TT) + t0;  // x[b][d][t0+m]

    // ---- stage z tile: zbuf[m][d] = x[b][d][t0+m]  (coalesced float4 over t)
    for (int i = tid; i < DD * (MT / 4); i += 256) {
        int m4 = i & (MT / 4 - 1);
        int d  = i >> 4;
        float4 v = *(const float4*)(xb + (size_t)d * TT + 4 * m4);
        int m = 4 * m4;
        zbuf[(m + 0) * ZS + d] = v.x;
        zbuf[(m + 1) * ZS + d] = v.y;
        zbuf[(m + 2) * ZS + d] = v.z;
        zbuf[(m + 3) * ZS + d] = v.w;
    }
    __syncthreads();

    // ---- initial row norms + loss accumulators
    if (tid < MT) {
        const float* zr = zbuf + tid * ZS;
        float s = 0.f;
#pragma unroll 8
        for (int d = 0; d < DD; ++d) s += zr[d] * zr[d];
        rn[tid]      = s;
        lossrow[tid] = 0.f;
    }
    __syncthreads();

    for (int c = 0; c < NC; ++c) {
        const float* E   = cb + (size_t)c * KK * DD;
        const float* eec = ee + c * KK;

        // running per-row best over this wave's 128 columns
        float bestv[2][8];
        int   besti[2][8];
#pragma unroll
        for (int j = 0; j < 2; ++j)
#pragma unroll
            for (int v = 0; v < 8; ++v) { bestv[j][v] = 3.4e38f; besti[j][v] = 0; }

        // A-frag LDS address: row m0+ln16 (+16 for second m-block),
        // K-pair (d0+2*hi, d0+2*hi+1)
        const int zoff0 = (m0 + ln16) * ZS + 2 * hi;

        for (int kt = 0; kt < 4; ++kt) {
            const int col0 = kq * 128 + kt * 32 + ln16;       // first k-tile col
            const float* Ep0 = E + (size_t)col0 * DD + 2 * hi;
            v8f acc[2][2];
#pragma unroll
            for (int j = 0; j < 2; ++j)
#pragma unroll
                for (int k = 0; k < 2; ++k)
                    acc[j][k] = (v8f){0.f, 0.f, 0.f, 0.f, 0.f, 0.f, 0.f, 0.f};

#pragma unroll 4
            for (int d0 = 0; d0 < DD; d0 += 4) {
                v2f a0 = *(const v2f*)(zbuf + zoff0 + d0);            // m0..+15
                v2f a1 = *(const v2f*)(zbuf + zoff0 + 16 * ZS + d0);  // m0+16..+31
                v2f b0 = *(const v2f*)(Ep0 + d0);                     // col0 tile
                v2f b1 = *(const v2f*)(Ep0 + 16 * DD + d0);           // col0+16 tile
                acc[0][0] = wmma_f32(a0, b0, acc[0][0]);
                acc[0][1] = wmma_f32(a0, b1, acc[0][1]);
                acc[1][0] = wmma_f32(a1, b0, acc[1][0]);
                acc[1][1] = wmma_f32(a1, b1, acc[1][1]);
            }

            const float ee0 = eec[col0];
            const float ee1 = eec[col0 + 16];
#pragma unroll
            for (int j = 0; j < 2; ++j) {
#pragma unroll
                for (int v = 0; v < 8; ++v) {
                    // score s_k = ||e_k||^2 - 2 z.e_k (||z||^2 const per row)
                    float s0 = fmaf(-2.0f, acc[j][0][v], ee0);
                    if (s0 < bestv[j][v]) { bestv[j][v] = s0; besti[j][v] = col0; }
                    float s1 = fmaf(-2.0f, acc[j][1][v], ee1);
                    if (s1 < bestv[j][v]) { bestv[j][v] = s1; besti[j][v] = col0 + 16; }
                }
            }
        }

        // butterfly argmin across the 16 lanes holding one row's columns
#pragma unroll
        for (int off = 1; off <= 8; off <<= 1) {
#pragma unroll
            for (int j = 0; j < 2; ++j) {
#pragma unroll
                for (int v = 0; v < 8; ++v) {
                    float ov = __shfl_xor(bestv[j][v], off, 32);
                    int   oi = __shfl_xor(besti[j][v], off, 32);
                    if (ov < bestv[j][v] ||
                        (ov == bestv[j][v] && oi < besti[j][v])) {
                        bestv[j][v] = ov; besti[j][v] = oi;
                    }
                }
            }
        }
        if (ln16 == 0) {
#pragma unroll
            for (int j = 0; j < 2; ++j) {
#pragma unroll
                for (int v = 0; v < 8; ++v) {
                    int row = m0 + j * 16 + v + 8 * hi;
                    pmin[row * 4 + kq] = bestv[j][v];
                    pidx[row * 4 + kq] = besti[j][v];
                }
            }
        }
        __syncthreads();

        // combine the four k-quarters (ascending cols; strict < keeps first min)
        if (tid < MT) {
            float smin = pmin[tid * 4 + 0];
            int   kmin = pidx[tid * 4 + 0];
#pragma unroll
            for (int q = 1; q < 4; ++q) {
                float v = pmin[tid * 4 + q];
                int   i = pidx[tid * 4 + q];
                if (v < smin) { smin = v; kmin = i; }
            }
            float nrm = rn[tid] + smin;   // = min dist = ||z - e||^2 = new ||z||^2
            rn[tid]       = nrm;
            lossrow[tid] += nrm;
            idxrow[tid]   = kmin;
            if (c == NC - 1) out_idx[mbase + tid] = (float)kmin;
        }
        __syncthreads();

        // residual update z -= E[idx] (not needed after last codebook)
        if (c < NC - 1) {
            for (int m = 0; m < MT; ++m) {
                const float* er = E + (size_t)idxrow[m] * DD;
                zbuf[m * ZS + tid] -= er[tid];       // tid == d, coalesced
            }
            __syncthreads();
        }
    }

    if (tid == 0) {
        float s = 0.f;
        for (int m = 0; m < MT; ++m) s += lossrow[m];
        partial[blockIdx.x] = s;
    }
}

// ---------------------------------------------------------------------------
// Kernel 3: deterministic reduction of per-block loss partials -> two scalars
// ---------------------------------------------------------------------------
__global__ __launch_bounds__(256) void rvq_reduce_kernel(const float* __restrict__ partial,
                                                         float* __restrict__ out) {
    __shared__ float sh[256];
    float s = 0.f;
    for (int i = threadIdx.x; i < NBLK; i += 256) s += partial[i];
    sh[threadIdx.x] = s;
    __syncthreads();
    for (int off = 128; off > 0; off >>= 1) {
        if (threadIdx.x < off) sh[threadIdx.x] += sh[threadIdx.x + off];
        __syncthreads();
    }
    if (threadIdx.x == 0) {
        float loss = sh[0] * (1.0f / ((float)NN * (float)DD));
        out[NN + 0] = loss;   // codebook_losses
        out[NN + 1] = loss;   // commitment_losses (numerically identical)
    }
}

// ---------------------------------------------------------------------------
extern "C" void kernel_launch(void* const* d_in, const int* in_sizes, int n_in,
                              void* d_out, int out_size, void* d_ws, size_t ws_size,
                              hipStream_t stream) {
    const float* x  = (const float*)d_in[0];   // [16, 256, 4096]
    const float* cb = (const float*)d_in[1];   // [3, 512, 256]
    float* out = (float*)d_out;                // [65536 idx][2 losses]

    float* ws      = (float*)d_ws;
    float* ee      = ws;                       // NC*KK floats
    float* partial = ws + NC * KK;             // NBLK floats

    const size_t smem = (size_t)MT * ZS * sizeof(float)   // z tile
                      + (size_t)MT * 3 * sizeof(float)    // rn, lossrow, idxrow
                      + (size_t)MT * 4 * sizeof(float)    // pmin
                      + (size_t)MT * 4 * sizeof(int);     // pidx

    hipFuncSetAttribute((const void*)rvq_main_kernel,
                        hipFuncAttributeMaxDynamicSharedMemorySize, (int)smem);

    rvq_ee_kernel<<<(NC * KK + 255) / 256, 256, 0, stream>>>(cb, ee);
    rvq_main_kernel<<<NBLK, 256, smem, stream>>>(x, cb, ee, out, partial);
    rvq_reduce_kernel<<<1, 256, 0, stream>>>(partial, out);
}